// AEVComputer_19585050869983
// MI455X (gfx1250) — compile-verified
//
#include <hip/hip_runtime.h>
#include <math.h>

#define RCR 5.2f
#define RCA 3.5f
#define PI_F 3.14159265358979323846f

typedef float v2f __attribute__((ext_vector_type(2)));
typedef float v8f __attribute__((ext_vector_type(8)));

__global__ __launch_bounds__(256) void aev_kernel(
    const float* __restrict__ coords,
    const float* __restrict__ EtaR, const float* __restrict__ ShfR,
    const float* __restrict__ EtaA, const float* __restrict__ Zeta,
    const float* __restrict__ ShfA, const float* __restrict__ ShfZ,
    const int* __restrict__ species,
    float* __restrict__ out)
{
    __shared__ float sh_c[96];          // batch coords (32 x 3)
    __shared__ int   sh_sp[32];         // batch species
    __shared__ float sShfR[16], sCz[8], sSz[8], sShfA[4];
    __shared__ float sEtaR, sEtaA, sZeta;
    __shared__ float rv[8][32 * 4];     // per-wave rvec, padded to 4
    __shared__ float dvs[8][32];        // per-wave distances
    __shared__ float fcm[8][32];        // per-wave masked angular cutoff fn
    __shared__ float dotsm[8][32 * 32]; // per-wave Gram matrix (WMMA output)
    __shared__ float racc[8][64];       // radial accumulators
    __shared__ float aacc[8][320];      // angular accumulators

    const int tid  = threadIdx.x;
    const int w    = tid >> 5;          // wave id 0..7
    const int lane = tid & 31;
    const int b    = blockIdx.x >> 2;   // batch
    const int i    = ((blockIdx.x & 3) << 3) + w; // atom center 0..31

    __builtin_prefetch(coords + (size_t)b * 96, 0, 1); // global_prefetch_b8

    // ---- cooperative block-shared loads ----
    if (tid < 96) sh_c[tid] = coords[(size_t)b * 96 + tid];
    if (tid < 32) sh_sp[tid] = species[b * 32 + tid];
    if (tid < 16) sShfR[tid] = ShfR[tid];
    if (tid < 8)  { float z = ShfZ[tid]; sCz[tid] = cosf(z); sSz[tid] = sinf(z); }
    if (tid < 4)  sShfA[tid] = ShfA[tid];
    if (tid == 0) { sEtaR = EtaR[0]; sEtaA = EtaA[0]; sZeta = Zeta[0]; }
    __syncthreads();

    // ---- zero wave-private accumulators ----
    racc[w][lane * 2]     = 0.f;
    racc[w][lane * 2 + 1] = 0.f;
#pragma unroll
    for (int q = 0; q < 10; ++q) aacc[w][q * 32 + lane] = 0.f;

    // ---- per-lane neighbor j = lane: rvec, distance, masks ----
    const int j = lane;
    const float cx = sh_c[i * 3 + 0], cy = sh_c[i * 3 + 1], cz = sh_c[i * 3 + 2];
    const float vx = sh_c[j * 3 + 0] - cx;
    const float vy = sh_c[j * 3 + 1] - cy;
    const float vz = sh_c[j * 3 + 2] - cz;
    rv[w][j * 4 + 0] = vx; rv[w][j * 4 + 1] = vy;
    rv[w][j * 4 + 2] = vz; rv[w][j * 4 + 3] = 0.f;  // K=4 pad for WMMA

    const float d2 = vx * vx + vy * vy + vz * vz;
    const float d  = sqrtf(d2 > 0.f ? d2 : 1.f);
    dvs[w][j] = d;

    const int  spi_raw = sh_sp[i];
    const int  spj_raw = sh_sp[j];
    const bool pair_ok = (spi_raw >= 0) && (spj_raw >= 0) && (j != i);
    const int  spj = min(max(spj_raw, 0), 3);

    // angular cutoff, pre-masked (zero kills the whole term later)
    const float fcA  = 0.5f * __cosf(d * (PI_F / RCA)) + 0.5f;
    const float fcmj = (pair_ok && d <= RCA) ? fcA : 0.f;
    fcm[w][j] = fcmj;

    // ---- radial features (scatter by species via LDS atomics) ----
    {
        const float fcR = 0.5f * __cosf(d * (PI_F / RCR)) + 0.5f;
        const float er  = sEtaR;
        if (pair_ok && d <= RCR) {
#pragma unroll
            for (int r = 0; r < 16; ++r) {
                float x = d - sShfR[r];
                atomicAdd(&racc[w][spj * 16 + r], 0.25f * __expf(-er * x * x) * fcR);
            }
        }
    }

    // ---- 32x32 Gram matrix dots[j][k] = rvec_j . rvec_k via V_WMMA_F32_16X16X4_F32 ----
    // f32 16x4 A layout: lane L holds (K = 2*(L>=16), K+1) of row (L&15) + 16*tile.
    // B (4x16) layout mirrors it, so the same operand registers serve A and B.
    const int half = lane >> 4;
    const int rsel = lane & 15;
    const int kb   = half << 1;
#if __has_builtin(__builtin_amdgcn_wmma_f32_16x16x4_f32)
    {
        v2f op[2];
        op[0].x = rv[w][(rsel) * 4 + kb];      op[0].y = rv[w][(rsel) * 4 + kb + 1];
        op[1].x = rv[w][(16 + rsel) * 4 + kb]; op[1].y = rv[w][(16 + rsel) * 4 + kb + 1];
#pragma unroll
        for (int mt = 0; mt < 2; ++mt) {
#pragma unroll
            for (int nt = 0; nt < 2; ++nt) {
                v8f acc = {};
                acc = __builtin_amdgcn_wmma_f32_16x16x4_f32(
                    false, op[mt], false, op[nt], (short)0, acc, false, false);
                // D layout: VGPR v, lanes 0-15 -> M=v, lanes 16-31 -> M=v+8; N = lane&15
#pragma unroll
                for (int v = 0; v < 8; ++v) {
                    dotsm[w][(mt * 16 + v + half * 8) * 32 + nt * 16 + rsel] = acc[v];
                }
            }
        }
    }
#else
    {
#pragma unroll 1
        for (int k = 0; k < 32; ++k)
            dotsm[w][j * 32 + k] =
                vx * rv[w][k * 4] + vy * rv[w][k * 4 + 1] + vz * rv[w][k * 4 + 2];
    }
#endif

    // ---- angular features over unordered pairs (j,k): circular-offset enumeration ----
    const float etaA = sEtaA, zeta = sZeta;
#pragma unroll 1
    for (int o = 1; o <= 16; ++o) {
        if (o == 16 && lane >= 16) break;       // o=16 pairs counted once
        const int k = (lane + o) & 31;
        const float fpr = fcmj * fcm[w][k];     // fc_j*fc_k, zero if either masked
        if (fpr > 0.f) {
            const float dk   = dvs[w][k];
            const float dot  = dotsm[w][lane * 32 + k];
            const float ca   = dot / fmaxf(d * dk, 1e-10f);
            const float c    = 0.95f * ca;                 // cos(theta)
            const float sc   = sqrtf(fmaxf(1.f - c * c, 0.f)); // sin(theta), theta in [0,pi]
            const float dmean = 0.5f * (d + dk);
            float f2v[4];
#pragma unroll
            for (int p = 0; p < 4; ++p) {
                float x = dmean - sShfA[p];
                f2v[p] = __expf(-etaA * x * x);
            }
            const int spk = min(max(sh_sp[k], 0), 3);
            const int a   = min(spj, spk);
            const int bb  = max(spj, spk);
            const int pidx = a * 4 - ((a * (a - 1)) >> 1) + (bb - a); // triu index
            float* ab = &aacc[w][pidx * 32];
#pragma unroll
            for (int t = 0; t < 8; ++t) {
                // cos(theta - ShfZ_t) = c*cos(s) + sin(theta)*sin(s)  (no arccos needed)
                float ct = fmaf(c, sCz[t], sc * sSz[t]);
                float f1 = __powf(0.5f * (1.f + ct), zeta);
                float s2 = 2.f * f1 * fpr;
#pragma unroll
                for (int p = 0; p < 4; ++p)
                    atomicAdd(&ab[p * 8 + t], s2 * f2v[p]);
            }
        }
    }

    // ---- writeout: 64 radial + 320 angular per center (coalesced) ----
    const size_t obase = ((size_t)(b * 32 + i)) * 384;
    out[obase + lane * 2]     = racc[w][lane * 2];
    out[obase + lane * 2 + 1] = racc[w][lane * 2 + 1];
#pragma unroll
    for (int q = 0; q < 10; ++q)
        out[obase + 64 + q * 32 + lane] = aacc[w][q * 32 + lane];
}

extern "C" void kernel_launch(void* const* d_in, const int* in_sizes, int n_in,
                              void* d_out, int out_size, void* d_ws, size_t ws_size,
                              hipStream_t stream) {
    const float* coords = (const float*)d_in[0];
    const float* EtaR   = (const float*)d_in[1];
    const float* ShfR   = (const float*)d_in[2];
    const float* EtaA   = (const float*)d_in[3];
    const float* Zeta   = (const float*)d_in[4];
    const float* ShfA   = (const float*)d_in[5];
    const float* ShfZ   = (const float*)d_in[6];
    const int*   species = (const int*)d_in[7];
    float* out = (float*)d_out;

    const int B = in_sizes[0] / 96;  // coordinates: B x 32 x 3
    dim3 grid(B * 4), block(256);    // 8 waves/block = 8 atom centers
    hipLaunchKernelGGL(aev_kernel, grid, block, 0, stream,
                       coords, EtaR, ShfR, EtaA, Zeta, ShfA, ShfZ, species, out);
}